// LineDescriptorHead_64269890617652
// MI455X (gfx1250) — compile-verified
//
#include <hip/hip_runtime.h>

typedef __attribute__((ext_vector_type(2))) float v2f;
typedef __attribute__((ext_vector_type(8))) float v8f;

#define CH_STRIDE (512 * 512)
#define LDA_F 194   // fused A stride (192 cols + pad); even -> 8B aligned, mod64=2 -> conflict free
#define LDA_H 66    // 64-col tiles stride

// One 16x16 f32 output tile of A(16xK, LDS) * B(Kx16, global), via V_WMMA_F32_16X16X4_F32.
// A layout per ISA: lanes 0-15 hold {K=4s, 4s+1} of row M=lane; lanes 16-31 hold {K=4s+2, 4s+3} of row M=lane-16.
// B layout per ISA: lanes index N; VGPR0/1 hold rows K=4s(+2), 4s+1(+3).
__device__ __forceinline__ v8f wmma_tile_f32(const float* __restrict__ A, int lda,
                                             const float* __restrict__ B, int ldb,
                                             int K, int lane) {
  v8f c = {0.f, 0.f, 0.f, 0.f, 0.f, 0.f, 0.f, 0.f};
  const int row   = lane & 15;
  const int khalf = (lane >> 4) << 1;   // 0 or 2
  const int col   = lane & 15;
  for (int s = 0; s < K; s += 4) {
    const int kb = s + khalf;
    v2f a, b;
    a[0] = A[row * lda + kb];
    a[1] = A[row * lda + kb + 1];
    b[0] = B[kb * ldb + col];
    b[1] = B[(kb + 1) * ldb + col];
    c = __builtin_amdgcn_wmma_f32_16x16x4_f32(false, a, false, b, (short)0, c, false, false);
  }
  return c;
}

__global__ __launch_bounds__(128)
void line_desc_kernel(const float* __restrict__ dmap,
                      const float* __restrict__ lines,
                      const float* __restrict__ wg1, const float* __restrict__ bg1,
                      const float* __restrict__ wg2, const float* __restrict__ bg2,
                      const float* __restrict__ wa1, const float* __restrict__ ba1,
                      const float* __restrict__ wa2, const float* __restrict__ ba2,
                      float* __restrict__ out, int nlines) {
  __shared__ float fusedA[16 * LDA_F];  // [16][192]: mean | max | g
  __shared__ float hid1S[16 * LDA_H];   // geo layer-1 out (A of GEMM1)
  __shared__ float hid2S[16 * LDA_H];   // head layer-1 out (A of GEMM3)
  __shared__ float outS[16 * LDA_H];    // head layer-2 out (pre-norm)
  __shared__ float geoS[16 * 8];
  __shared__ float lineS[16 * 4];
  __shared__ float normS[16];

  const int tid  = threadIdx.x;
  const int lane = tid & 31;
  const int wv   = tid >> 5;            // wave id: owns output cols [16*wv, 16*wv+16)
  const int blk0 = blockIdx.x * 16;

  // ---- Stage 1: per-line geometry (16 threads) ----
  if (tid < 16) {
    const int ln = blk0 + tid;
    float l0 = 0.f, l1 = 0.f, l2 = 0.f, l3 = 0.f;
    if (ln < nlines) {
      const float4 v = ((const float4*)lines)[ln];
      l0 = v.x; l1 = v.y; l2 = v.z; l3 = v.w;
    }
    const float dx = fabsf(l0 - l2), dy = fabsf(l1 - l3);
    const bool swap = (dx > dy) ? (l0 > l2) : (l1 > l3);
    const float sx = swap ? l2 : l0, sy = swap ? l3 : l1;
    const float ex = swap ? l0 : l2, ey = swap ? l1 : l3;
    const float cx = (sx + ex) * 0.5f, cy = (sy + ey) * 0.5f;
    const float ddx = ex - sx, ddy = ey - sy;
    const float L = fmaxf(sqrtf(ddx * ddx + ddy * ddy), 1e-6f);
    geoS[tid * 8 + 0] = cx * (1.f / 512.f);
    geoS[tid * 8 + 1] = cy * (1.f / 512.f);
    geoS[tid * 8 + 2] = ddx / L;
    geoS[tid * 8 + 3] = ddy / L;
    geoS[tid * 8 + 4] = L * (1.f / 724.0773439350246f);   // L / (512*sqrt(2))
    lineS[tid * 4 + 0] = sx; lineS[tid * 4 + 1] = sy;
    lineS[tid * 4 + 2] = ex; lineS[tid * 4 + 3] = ey;
  }
  __syncthreads();

  // ---- Stage 2: bilinear gather, mean/max over 8 points (all 128 threads) ----
  {
    const int l  = tid >> 3;          // line within block
    const int c0 = (tid & 7) * 8;     // 8-channel slice
    const float sx = lineS[l * 4 + 0], sy = lineS[l * 4 + 1];
    const float ex = lineS[l * 4 + 2], ey = lineS[l * 4 + 3];
    float sum[8], mx[8];
#pragma unroll
    for (int i = 0; i < 8; i++) { sum[i] = 0.f; mx[i] = -3.4e38f; }
    const float tv[8] = {0.f, 0.5f, 1.f, 1.f/6.f, 2.f/6.f, 3.f/6.f, 4.f/6.f, 5.f/6.f};
    for (int p = 0; p < 8; p++) {
      const float t = tv[p];
      const float x = fminf(fmaxf(sx + (ex - sx) * t, 0.f), 511.f);
      const float y = fminf(fmaxf(sy + (ey - sy) * t, 0.f), 511.f);
      const float x0f = floorf(x), y0f = floorf(y);
      const float wx = x - x0f, wy = y - y0f;
      const int x0 = (int)x0f, y0 = (int)y0f;
      const int x1 = min(x0 + 1, 511), y1 = min(y0 + 1, 511);
      const float w00 = (1.f - wx) * (1.f - wy), w10 = wx * (1.f - wy);
      const float w01 = (1.f - wx) * wy,         w11 = wx * wy;
      const int o00 = y0 * 512 + x0, o10 = y0 * 512 + x1;
      const int o01 = y1 * 512 + x0, o11 = y1 * 512 + x1;
      const float* bp = dmap + (size_t)c0 * CH_STRIDE;
#pragma unroll
      for (int i = 0; i < 8; i++) {
        const float v = bp[o00] * w00 + bp[o10] * w10 + bp[o01] * w01 + bp[o11] * w11;
        sum[i] += v;
        mx[i] = fmaxf(mx[i], v);
        bp += CH_STRIDE;
      }
    }
#pragma unroll
    for (int i = 0; i < 8; i++) {
      fusedA[l * LDA_F + c0 + i]      = sum[i] * 0.125f;  // mean over 8 points
      fusedA[l * LDA_F + 64 + c0 + i] = mx[i];            // max
    }
  }

  // ---- Stage 3: geo MLP layer 1 (K=5, scalar) ----
  {
    const int l  = tid >> 3;
    const int h0 = (tid & 7) * 8;
#pragma unroll
    for (int j = 0; j < 8; j++) {
      const int h = h0 + j;
      float acc = bg1[h];
#pragma unroll
      for (int k = 0; k < 5; k++) acc += geoS[l * 8 + k] * wg1[k * 64 + h];
      hid1S[l * LDA_H + h] = fmaxf(acc, 0.f);
    }
  }
  __syncthreads();

  // ---- GEMM1: g = relu(hid1(16x64) @ wg2(64x64) + bg2) -> fused cols 128..191 ----
  {
    const v8f c = wmma_tile_f32(hid1S, LDA_H, wg2 + wv * 16, 64, 64, lane);
    const int n = lane & 15, mb = (lane >> 4) * 8;
    const float bias = bg2[wv * 16 + n];
#pragma unroll
    for (int v = 0; v < 8; v++)
      fusedA[(mb + v) * LDA_F + 128 + wv * 16 + n] = fmaxf(c[v] + bias, 0.f);
  }
  __syncthreads();

  // ---- GEMM2: h = relu(fused(16x192) @ wa1(192x64) + ba1) ----
  {
    const v8f c = wmma_tile_f32(fusedA, LDA_F, wa1 + wv * 16, 64, 192, lane);
    const int n = lane & 15, mb = (lane >> 4) * 8;
    const float bias = ba1[wv * 16 + n];
#pragma unroll
    for (int v = 0; v < 8; v++)
      hid2S[(mb + v) * LDA_H + wv * 16 + n] = fmaxf(c[v] + bias, 0.f);
  }
  __syncthreads();

  // ---- GEMM3: o = h(16x64) @ wa2(64x64) + ba2 ----
  {
    const v8f c = wmma_tile_f32(hid2S, LDA_H, wa2 + wv * 16, 64, 64, lane);
    const int n = lane & 15, mb = (lane >> 4) * 8;
    const float bias = ba2[wv * 16 + n];
#pragma unroll
    for (int v = 0; v < 8; v++)
      outS[(mb + v) * LDA_H + wv * 16 + n] = c[v] + bias;
  }
  __syncthreads();

  // ---- L2 norm per line ----
  if (tid < 16) {
    float s = 0.f;
#pragma unroll 8
    for (int c = 0; c < 64; c++) { const float v = outS[tid * LDA_H + c]; s += v * v; }
    normS[tid] = 1.f / fmaxf(sqrtf(s), 1e-12f);
  }
  __syncthreads();

  // ---- Store normalized descriptors ----
  {
    const int l = tid >> 3, c0 = (tid & 7) * 8;
    const int row = blk0 + l;
    if (row < nlines) {
      const float nv = normS[l];
#pragma unroll
      for (int j = 0; j < 8; j++)
        out[(size_t)row * 64 + c0 + j] = outS[l * LDA_H + c0 + j] * nv;
    }
  }
}

extern "C" void kernel_launch(void* const* d_in, const int* in_sizes, int n_in,
                              void* d_out, int out_size, void* d_ws, size_t ws_size,
                              hipStream_t stream) {
  const float* dmap = (const float*)d_in[0];
  const float* lines = (const float*)d_in[1];
  const float* wg1 = (const float*)d_in[2];
  const float* bg1 = (const float*)d_in[3];
  const float* wg2 = (const float*)d_in[4];
  const float* bg2 = (const float*)d_in[5];
  const float* wa1 = (const float*)d_in[6];
  const float* ba1 = (const float*)d_in[7];
  const float* wa2 = (const float*)d_in[8];
  const float* ba2 = (const float*)d_in[9];
  float* out = (float*)d_out;

  const int nlines = in_sizes[1] / 4;      // lines is (N,4)
  const int nblk = (nlines + 15) / 16;
  hipLaunchKernelGGL(line_desc_kernel, dim3(nblk), dim3(128), 0, stream,
                     dmap, lines, wg1, bg1, wg2, bg2, wa1, ba1, wa2, ba2, out, nlines);
}